// HomoHingeLoss_20547123544860
// MI455X (gfx1250) — compile-verified
//
#include <hip/hip_runtime.h>

// Problem constants (from reference setup_inputs)
#define NB    8
#define C_DIM 256
#define HR    40
#define WR    40
#define HW    (HR * WR)            // 1600
#define GS    8
#define HFULL (HR * GS)            // 320
#define S_ELEMS ((long long)NB * HW * HW)  // 20,480,000

typedef __attribute__((ext_vector_type(2))) float v2f;
typedef __attribute__((ext_vector_type(8))) float v8f;

// ---------------------------------------------------------------------------
// Kernel 1: reduced mask  r_mask[n,i,j] = prod over 8x8 block of mask[n,0,:,:]
// ---------------------------------------------------------------------------
__global__ void rmask_kernel(const float* __restrict__ mask,
                             float* __restrict__ rmask) {
  int t = blockIdx.x * blockDim.x + threadIdx.x;
  if (t >= NB * HR * WR) return;
  int j = t % WR;
  int i = (t / WR) % HR;
  int n = t / (HR * WR);
  const float* mp = mask + (size_t)n * HFULL * HFULL;
  float p = 1.0f;
  for (int dy = 0; dy < GS; ++dy) {
    const float* row = mp + (size_t)(i * GS + dy) * HFULL + j * GS;
#pragma unroll
    for (int dx = 0; dx < GS; ++dx) p *= row[dx];
  }
  rmask[t] = p;
}

// ---------------------------------------------------------------------------
// Kernel 2: s[n,h,w,i,j] = [dist(grid(h,w), warp(grid(i,j))) <= 7.5] * rmask[n,i,j]
// ---------------------------------------------------------------------------
__global__ void s_kernel(const float* __restrict__ homo,
                         const float* __restrict__ rmask,
                         float* __restrict__ s_out) {
  long long t = (long long)blockIdx.x * blockDim.x + threadIdx.x;
  if (t >= S_ELEMS) return;
  int j = (int)(t % WR);
  long long r = t / WR;
  int i = (int)(r % HR); r /= HR;
  int w = (int)(r % WR); r /= WR;
  int h = (int)(r % HR);
  int n = (int)(r / HR);

  // warped cell center for (i, j):  p' = H @ [x, y, 1]
  float gx = (float)(j * GS + GS / 2);
  float gy = (float)(i * GS + GS / 2);
  float wx = homo[0] * gx + homo[1] * gy + homo[2];
  float wy = homo[3] * gx + homo[4] * gy + homo[5];
  float wz = homo[6] * gx + homo[7] * gy + homo[8];
  float inv = 1.0f / wz;
  wx *= inv; wy *= inv;

  float dx = (float)(w * GS + GS / 2) - wx;
  float dy = (float)(h * GS + GS / 2) - wy;
  const float thr = (GS - 0.5f) * (GS - 0.5f);   // 7.5^2
  float ind = (dx * dx + dy * dy <= thr) ? 1.0f : 0.0f;
  s_out[t] = ind * rmask[(n * HR + i) * WR + j];
}

// ---------------------------------------------------------------------------
// Kernel 3: dot_des[n,p,q] = sum_c des1[n,c,p] * des2[n,c,q]
// fp32 WMMA GEMM: each wave owns a 16(M)x64(N) strip (4 accumulators, A reuse),
// 4 waves/block -> 64x64 tile, grid 25x25x8, K loop 256 in steps of 4.
// ---------------------------------------------------------------------------
__global__ void __launch_bounds__(128)
gemm_wmma_kernel(const float* __restrict__ des1,
                 const float* __restrict__ des2,
                 float* __restrict__ dot_out) {
  const int lane = threadIdx.x & 31;
  const int wave = threadIdx.x >> 5;
  const int n  = blockIdx.z;
  const int pm = (blockIdx.y * 4 + wave) * 16;   // M tile base (rows p)
  const int qn = blockIdx.x * 64;                // N tile base (cols q)

  const float* A = des1 + (size_t)n * C_DIM * HW;  // [c][p]
  const float* B = des2 + (size_t)n * C_DIM * HW;  // [c][q]

  // fp32 16x16x4 fragment layout: lane holds m(or n)=lane%16, K = 2*(lane/16)+v
  const int row  = lane & 15;
  const int koff = (lane >> 4) << 1;

  v8f acc0 = {}, acc1 = {}, acc2 = {}, acc3 = {};

#pragma unroll 2
  for (int kk = 0; kk < C_DIM; kk += 4) {
    const float* ap = A + (size_t)(kk + koff) * HW + pm + row;
    v2f a; a.x = ap[0]; a.y = ap[HW];

    const float* bp = B + (size_t)(kk + koff) * HW + qn + row;
    v2f b0; b0.x = bp[0];  b0.y = bp[HW];
    v2f b1; b1.x = bp[16]; b1.y = bp[HW + 16];
    v2f b2; b2.x = bp[32]; b2.y = bp[HW + 32];
    v2f b3; b3.x = bp[48]; b3.y = bp[HW + 48];

    acc0 = __builtin_amdgcn_wmma_f32_16x16x4_f32(false, a, false, b0, (short)0, acc0, false, false);
    acc1 = __builtin_amdgcn_wmma_f32_16x16x4_f32(false, a, false, b1, (short)0, acc1, false, false);
    acc2 = __builtin_amdgcn_wmma_f32_16x16x4_f32(false, a, false, b2, (short)0, acc2, false, false);
    acc3 = __builtin_amdgcn_wmma_f32_16x16x4_f32(false, a, false, b3, (short)0, acc3, false, false);
  }

  // C/D layout: VGPR r, lanes 0-15 -> M=r, N=lane; lanes 16-31 -> M=r+8, N=lane-16
  float* D = dot_out + (size_t)n * HW * HW;
  const int mbase = pm + ((lane >> 4) << 3);
  const int ncol  = lane & 15;
#pragma unroll
  for (int r2 = 0; r2 < 8; ++r2) {
    float* drow = D + (size_t)(mbase + r2) * HW + qn + ncol;
    drow[0]  = acc0[r2];
    drow[16] = acc1[r2];
    drow[32] = acc2[r2];
    drow[48] = acc3[r2];
  }
}

// ---------------------------------------------------------------------------
// Kernel 4: per-element hinge terms + 3x3 dilation over (i,j); block-reduced
// deterministic partial sums.
// ---------------------------------------------------------------------------
__global__ void __launch_bounds__(256)
loss_kernel(const float* __restrict__ s_buf,
            const float* __restrict__ dot_buf,
            const float* __restrict__ rmask,
            float* __restrict__ partials) {
  __shared__ float red[256];
  const long long stride = (long long)gridDim.x * blockDim.x;
  float acc = 0.0f;

  for (long long t = (long long)blockIdx.x * blockDim.x + threadIdx.x;
       t < S_ELEMS; t += stride) {
    int j = (int)(t % WR);
    long long r = t / WR;
    int i = (int)(r % HR);
    long long slab = t - (long long)(i * WR + j);   // [n][h][w][0][0]
    int n = (int)(t / ((long long)HW * HW));

    float sc = s_buf[t];

    // 3x3 ones-conv over (i,j) with zero (SAME) padding, then clip to [0,1]
    float dsum = 0.0f;
#pragma unroll
    for (int di = -1; di <= 1; ++di) {
      int ii = i + di;
      if (ii < 0 || ii >= HR) continue;
#pragma unroll
      for (int dj = -1; dj <= 1; ++dj) {
        int jj = j + dj;
        if (jj < 0 || jj >= WR) continue;
        dsum += s_buf[slab + ii * WR + jj];
      }
    }
    float dil = fminf(fmaxf(dsum, 0.0f), 1.0f);
    float rm  = rmask[(n * HR + i) * WR + j];
    float ns  = (dil - sc) * rm;

    float dot = dot_buf[t];
    float pos = fmaxf(1.0f - dot, 0.0f);   // POS_MARGIN = 1.0
    float neg = fmaxf(dot - 0.2f, 0.0f);   // NEG_MARGIN = 0.2
    acc += 250.0f * sc * pos + ns * neg;   // POS_LAMBDA = 250
  }

  red[threadIdx.x] = acc;
  __syncthreads();
  for (int off = 128; off > 0; off >>= 1) {
    if (threadIdx.x < off) red[threadIdx.x] += red[threadIdx.x + off];
    __syncthreads();
  }
  if (threadIdx.x == 0) partials[blockIdx.x] = red[0];
}

__global__ void __launch_bounds__(256)
final_reduce_kernel(const float* __restrict__ partials, int np,
                    float* __restrict__ out_loss) {
  __shared__ float red[256];
  float acc = 0.0f;
  for (int t = threadIdx.x; t < np; t += 256) acc += partials[t];
  red[threadIdx.x] = acc;
  __syncthreads();
  for (int off = 128; off > 0; off >>= 1) {
    if (threadIdx.x < off) red[threadIdx.x] += red[threadIdx.x + off];
    __syncthreads();
  }
  if (threadIdx.x == 0) out_loss[0] = red[0] * (1.0f / (HW * 4));  // norm = hr*wr*4
}

// ---------------------------------------------------------------------------
// Launch
// ---------------------------------------------------------------------------
extern "C" void kernel_launch(void* const* d_in, const int* in_sizes, int n_in,
                              void* d_out, int out_size, void* d_ws, size_t ws_size,
                              hipStream_t stream) {
  const float* des1 = (const float*)d_in[0];
  const float* des2 = (const float*)d_in[1];
  const float* homo = (const float*)d_in[2];
  const float* mask = (const float*)d_in[3];

  float* out     = (float*)d_out;
  float* loss    = out;                       // [0]
  float* s_out   = out + 1;                   // [1 .. 1+S_ELEMS)
  float* dot_out = out + 1 + (size_t)S_ELEMS; // [1+S_ELEMS ..)

  float* rmask    = (float*)d_ws;             // 12800 floats
  float* partials = rmask + NB * HR * WR;     // 5000 floats

  // 1) reduced mask
  rmask_kernel<<<(NB * HR * WR + 255) / 256, 256, 0, stream>>>(mask, rmask);

  // 2) positive-match indicator tensor s
  s_kernel<<<(unsigned)(S_ELEMS / 256), 256, 0, stream>>>(homo, rmask, s_out);

  // 3) batched fp32 WMMA GEMM -> dot_des
  dim3 g(HW / 64, HW / 64, NB);               // 25 x 25 x 8
  gemm_wmma_kernel<<<g, 128, 0, stream>>>(des1, des2, dot_out);

  // 4) hinge loss partial sums (deterministic: fixed strides + tree reduce)
  const int nblocks = 5000;
  loss_kernel<<<nblocks, 256, 0, stream>>>(s_out, dot_out, rmask, partials);

  // 5) final deterministic reduce
  final_reduce_kernel<<<1, 256, 0, stream>>>(partials, nblocks, loss);
}